// Subtractor4Bit_23407571764120
// MI455X (gfx1250) — compile-verified
//
#include <hip/hip_runtime.h>

// 4-bit ripple-borrow subtractor over {0.0,1.0}-valued float tensors.
// Memory-bound streaming kernel: ~436 MB of traffic, ~19 us at 23.3 TB/s.
// Strategy: 128-bit loads/stores, non-temporal (TH_NT) cache policy on
// gfx1250 (inputs are 256 MiB > 192 MB L2, touched once), exact boolean
// logic instead of the float-polynomial soft gates (identical results for
// 0/1 inputs since _xor/_and/_or are exact there).

// Clang native vector type: required by __builtin_nontemporal_load/store
// (HIP's float4 is a class and is rejected by the builtin).
typedef float v4f __attribute__((ext_vector_type(4)));

__global__ __launch_bounds__(256) void
Subtractor4Bit_kernel(const v4f* __restrict__ A,
                      const v4f* __restrict__ B,
                      v4f*  __restrict__ R,        // result, (N,4)
                      float* __restrict__ Bout,    // borrow, (N,1)
                      int n_rows) {
    int i = blockIdx.x * blockDim.x + threadIdx.x;
    if (i >= n_rows) return;

    // Streaming 128-bit loads with non-temporal hint (gfx1250 th:TH_NT).
    v4f a = __builtin_nontemporal_load(&A[i]);
    v4f b = __builtin_nontemporal_load(&B[i]);

    // Inputs are exactly 0.0f or 1.0f -> booleanize.
    unsigned a0 = (a[0] != 0.0f), a1 = (a[1] != 0.0f), a2 = (a[2] != 0.0f), a3 = (a[3] != 0.0f);
    unsigned b0 = (b[0] != 0.0f), b1 = (b[1] != 0.0f), b2 = (b[2] != 0.0f), b3 = (b[3] != 0.0f);

    // Reference iterates i = 3,2,1,0 (bit 3 is LSB of the subtraction).
    //   diff   = a ^ b ^ borrow_in
    //   borrow = (~a & b) | (~a & bin) | (b & bin) = (~a & (b | bin)) | (b & bin)
    unsigned bw = 0u;
    unsigned d3 = a3 ^ b3 ^ bw;  bw = ((~a3) & (b3 | bw)) | (b3 & bw);
    unsigned d2 = a2 ^ b2 ^ bw;  bw = ((~a2) & (b2 | bw)) | (b2 & bw);
    unsigned d1 = a1 ^ b1 ^ bw;  bw = ((~a1) & (b1 | bw)) | (b1 & bw);
    unsigned d0 = a0 ^ b0 ^ bw;  bw = ((~a0) & (b0 | bw)) | (b0 & bw);

    v4f r;
    r[0] = (d0 & 1u) ? 1.0f : 0.0f;
    r[1] = (d1 & 1u) ? 1.0f : 0.0f;
    r[2] = (d2 & 1u) ? 1.0f : 0.0f;
    r[3] = (d3 & 1u) ? 1.0f : 0.0f;

    __builtin_nontemporal_store(r, &R[i]);
    __builtin_nontemporal_store((bw & 1u) ? 1.0f : 0.0f, &Bout[i]);
}

extern "C" void kernel_launch(void* const* d_in, const int* in_sizes, int n_in,
                              void* d_out, int out_size, void* d_ws, size_t ws_size,
                              hipStream_t stream) {
    (void)n_in; (void)d_ws; (void)ws_size; (void)out_size;

    const v4f* A = (const v4f*)d_in[0];         // (N,4) float32, rows are 16B-aligned
    const v4f* B = (const v4f*)d_in[1];
    const int n_rows = in_sizes[0] / 4;         // N

    v4f*   R    = (v4f*)d_out;                  // first 4N floats: result
    float* Bout = (float*)d_out + (size_t)4 * n_rows;  // last N floats: borrow

    const int block = 256;                      // 8 wave32s
    const int grid  = (n_rows + block - 1) / block;
    Subtractor4Bit_kernel<<<grid, block, 0, stream>>>(A, B, R, Bout, n_rows);
}